// NNUE_31525059952895
// MI455X (gfx1250) — compile-verified
//
#include <hip/hip_runtime.h>
#include <math.h>

// ---------------------------------------------------------------------------
// NNUE forward loss on MI455X (gfx1250).
//
// Roofline: the feature transformer streams 1.34 GB of fp32 features at
// 2 FLOP/byte -> pure HBM-bound (~58 us floor at 23.3 TB/s). Everything else
// is noise. So: stream features exactly once via V_WMMA_F32_16X16X4_F32,
// keep ft_w (1.3 MB) hot in L2 (shared across all 128 batch tiles), write
// deterministic per-slice partials (no atomics), fuse the tiny tail net.
//
// Stage 1 (nnue_ft_wmma): 128 batch tiles x 32 K-slices x 2 colors,
//   8 waves/block (8192 waves). Each wave: 640 fp32 WMMAs over a 2560-K
//   slice, one float2 load per lane per WMMA for A (features) and B (ft_w),
//   one WGP-scope prefetch per 64B line of per-lane stream.
// Stage 2 (nnue_tail): deterministic slice reduction + bias + clip +
//   L1(8x8) + L2(1x8) + sigmoid loss, one thread per sample.
// ---------------------------------------------------------------------------

typedef __attribute__((ext_vector_type(2))) float v2f;
typedef __attribute__((ext_vector_type(8))) float v8f;

#define NF        81920      // NUM_FEATURES
#define BATCH     2048
#define TILES     128        // BATCH / 16
#define SLICES    32         // K slices per color
#define SLICE_K   (NF / SLICES)   // 2560 K per slice -> 640 WMMAs per wave
#define WPB       8          // waves per block

// Workspace: partial[(z*SLICES + slice)][row 0..2047][n 0..3]  (2 MB fp32)

__global__ __launch_bounds__(256)
void nnue_ft_wmma(const float* __restrict__ whiteF,
                  const float* __restrict__ blackF,
                  const float* __restrict__ ftw,
                  float* __restrict__ partial)
{
    const int lane  = threadIdx.x & 31;
    const int wid   = threadIdx.x >> 5;
    const int tile  = blockIdx.x;                 // 0..127 batch tile
    const int slice = blockIdx.y * WPB + wid;     // 0..31 K slice
    const int zcol  = blockIdx.z;                 // 0 = white, 1 = black

    const float* feat = zcol ? blackF : whiteF;

    // fp32 WMMA A-matrix (16x4) lane mapping:
    //   lanes 0-15 : M=lane,    VGPR0=K0, VGPR1=K1
    //   lanes 16-31: M=lane-16, VGPR0=K2, VGPR1=K3
    // -> each lane loads one float2 of features and one float2 of weights.
    const int n     = lane & 15;          // B/D column (0..3 useful)
    const int half  = lane >> 4;          // 0 or 1
    const int kHalf = half * 2;           // K offset within 4-wide chunk
    const int nb    = n & 3;              // keep weight reads in-bounds;
                                          // cols 4..15 of D are never stored

    const size_t rowA  = (size_t)tile * 16 + n;
    const size_t kBase = (size_t)slice * SLICE_K + kHalf;

    const float* aPtr = feat + rowA * NF + kBase;
    const float* bPtr = ftw + (size_t)nb * NF + kBase;

    v8f c = {};
    // Per-lane stream advances 64 B (one cache line) per 16 K: one prefetch
    // per outer iteration, four WMMAs inside.
    for (int k = 0; k < SLICE_K; k += 16) {
        __builtin_prefetch(aPtr + k + 512, 0, 3);   // +2 KB, WGP scope
        #pragma unroll
        for (int kk = 0; kk < 16; kk += 4) {
            v2f a = *(const v2f*)(aPtr + k + kk);
            v2f b = *(const v2f*)(bPtr + k + kk);
            // D = A*B + C : 16x16x4 fp32 WMMA
            c = __builtin_amdgcn_wmma_f32_16x16x4_f32(
                    /*neg_a=*/false, a, /*neg_b=*/false, b,
                    /*c_mod=*/(short)0, c, /*reuse_a=*/false, /*reuse_b=*/false);
        }
    }

    // C/D layout: VGPR j holds (M = j + half*8, N = lane&15).
    // Store only the 4 real output columns of this wave's 16x4 partial tile.
    if (n < 4) {
        const int   rowBase = tile * 16 + half * 8;
        const size_t base   = (size_t)(zcol * SLICES + slice) * BATCH * 4;
        #pragma unroll
        for (int j = 0; j < 8; ++j)
            partial[base + (size_t)(rowBase + j) * 4 + n] = c[j];
    }
}

__global__ __launch_bounds__(256)
void nnue_tail(const float* __restrict__ partial,
               const float* __restrict__ turn,
               const float* __restrict__ score,
               const float* __restrict__ result,
               const float* __restrict__ ftb,
               const float* __restrict__ l1w,
               const float* __restrict__ l1b,
               const float* __restrict__ l2w,
               const float* __restrict__ l2b,
               float* __restrict__ out)
{
    const int i = blockIdx.x * blockDim.x + threadIdx.x;
    if (i >= BATCH) return;

    // Deterministic reduction over K-slices, then add FT bias.
    float w[4], b[4];
    #pragma unroll
    for (int m = 0; m < 4; ++m) { w[m] = ftb[m]; b[m] = ftb[m]; }
    for (int s = 0; s < SLICES; ++s) {
        const size_t baseW = ((size_t)s * BATCH + i) * 4;
        const size_t baseB = ((size_t)(SLICES + s) * BATCH + i) * 4;
        #pragma unroll
        for (int m = 0; m < 4; ++m) {
            w[m] += partial[baseW + m];
            b[m] += partial[baseB + m];
        }
    }

    // turn-select + clip to [0,1]
    const float t = turn[i];
    float x[8];
    #pragma unroll
    for (int m = 0; m < 4; ++m) {
        x[m]     = t * w[m] + (1.0f - t) * b[m];
        x[m + 4] = t * b[m] + (1.0f - t) * w[m];
    }
    #pragma unroll
    for (int m = 0; m < 8; ++m) x[m] = fminf(fmaxf(x[m], 0.0f), 1.0f);

    // L1: 8x8 + clip
    float y[8];
    #pragma unroll
    for (int j = 0; j < 8; ++j) {
        float acc = l1b[j];
        #pragma unroll
        for (int m = 0; m < 8; ++m) acc += l1w[j * 8 + m] * x[m];
        y[j] = fminf(fmaxf(acc, 0.0f), 1.0f);
    }

    // L2: 1x8
    float mr = l2b[0];
    #pragma unroll
    for (int j = 0; j < 8; ++j) mr += l2w[j] * y[j];

    // sigmoid loss, lambda = 0.5, scaling = 400
    const float wm = 1.0f / (1.0f + expf(-mr * (1.0f / 400.0f)));
    const float wt = 1.0f / (1.0f + expf(-score[i] * (1.0f / 400.0f)));
    const float de = wm - wt;
    const float dr = wm - result[i];
    out[i] = 0.5f * de * de + 0.5f * dr * dr;
}

extern "C" void kernel_launch(void* const* d_in, const int* in_sizes, int n_in,
                              void* d_out, int out_size, void* d_ws, size_t ws_size,
                              hipStream_t stream)
{
    const float* whiteF = (const float*)d_in[0];
    const float* blackF = (const float*)d_in[1];
    const float* turn   = (const float*)d_in[2];
    const float* score  = (const float*)d_in[3];
    const float* result = (const float*)d_in[4];
    const float* ftw    = (const float*)d_in[5];
    const float* ftb    = (const float*)d_in[6];
    const float* l1w    = (const float*)d_in[7];
    const float* l1b    = (const float*)d_in[8];
    const float* l2w    = (const float*)d_in[9];
    const float* l2b    = (const float*)d_in[10];

    float* partial = (float*)d_ws;   // needs 2*SLICES*BATCH*4 floats = 2 MB

    dim3 grid1(TILES, SLICES / WPB, 2);
    nnue_ft_wmma<<<grid1, 256, 0, stream>>>(whiteF, blackF, ftw, partial);

    nnue_tail<<<(BATCH + 255) / 256, 256, 0, stream>>>(
        partial, turn, score, result, ftb, l1w, l1b, l2w, l2b, (float*)d_out);
}